// COAPBodyModel_11527692223040
// MI455X (gfx1250) — compile-verified
//
#include <hip/hip_runtime.h>
#include <hip/hip_bf16.h>

#define KBOX 1024
#define EPS_F 1e-4f

typedef __attribute__((ext_vector_type(2))) float v2f;
typedef __attribute__((ext_vector_type(8))) float v8f;
typedef __attribute__((ext_vector_type(4))) unsigned int u32x4;
typedef __attribute__((ext_vector_type(8))) int i32x8;
typedef __attribute__((ext_vector_type(4))) int i32x4;

__device__ __forceinline__ unsigned uminu(unsigned a, unsigned b) { return a < b ? a : b; }

// interleave 16 bits into even positions of a 32-bit word
__device__ __forceinline__ unsigned spread16(unsigned x) {
    x &= 0xFFFFu;
    x = (x | (x << 8)) & 0x00FF00FFu;
    x = (x | (x << 4)) & 0x0F0F0F0Fu;
    x = (x | (x << 2)) & 0x33333333u;
    x = (x | (x << 1)) & 0x55555555u;
    return x;
}

// ---------------------------------------------------------------------------
// Phase 1: per box k, pose 8 corners into world space (write output 1 + ws),
// and build the analytic rigid inverse  inv(T) = [R^T | -R^T t ; 0 0 0 1].
// ---------------------------------------------------------------------------
__global__ void phase1_pose_and_invert(const float* __restrict__ bbmin,
                                       const float* __restrict__ bbmax,
                                       const float* __restrict__ T,
                                       float* __restrict__ btWS,      // K x 16
                                       float* __restrict__ hposedWS,  // K x 8 x 4
                                       float* __restrict__ outPosed)  // K x 8 x 3
{
    int k = blockIdx.x * blockDim.x + threadIdx.x;
    if (k >= KBOX) return;

    const float* Tk = T + k * 16;
    float R00 = Tk[0],  R01 = Tk[1],  R02 = Tk[2],  t0 = Tk[3];
    float R10 = Tk[4],  R11 = Tk[5],  R12 = Tk[6],  t1 = Tk[7];
    float R20 = Tk[8],  R21 = Tk[9],  R22 = Tk[10], t2 = Tk[11];

    float mnx = bbmin[k*3+0], mny = bbmin[k*3+1], mnz = bbmin[k*3+2];
    float mxx = bbmax[k*3+0], mxy = bbmax[k*3+1], mxz = bbmax[k*3+2];

#pragma unroll
    for (int c = 0; c < 8; ++c) {
        // _COMBOS order: bit2 -> x uses max, bit1 -> y, bit0 -> z
        float cx = (c & 4) ? mxx : mnx;
        float cy = (c & 2) ? mxy : mny;
        float cz = (c & 1) ? mxz : mnz;
        float px = R00*cx + R01*cy + R02*cz + t0;
        float py = R10*cx + R11*cy + R12*cz + t1;
        float pz = R20*cx + R21*cy + R22*cz + t2;
        float4 h = make_float4(px, py, pz, 1.0f);
        *(float4*)(hposedWS + (size_t)(k*8 + c) * 4) = h;
        outPosed[(size_t)(k*8 + c) * 3 + 0] = px;
        outPosed[(size_t)(k*8 + c) * 3 + 1] = py;
        outPosed[(size_t)(k*8 + c) * 3 + 2] = pz;
    }

    float* B = btWS + (size_t)k * 16;
    B[0]  = R00; B[1]  = R10; B[2]  = R20; B[3]  = -(R00*t0 + R10*t1 + R20*t2);
    B[4]  = R01; B[5]  = R11; B[6]  = R21; B[7]  = -(R01*t0 + R11*t1 + R21*t2);
    B[8]  = R02; B[9]  = R12; B[10] = R22; B[11] = -(R02*t0 + R12*t1 + R22*t2);
    B[12] = 0.f; B[13] = 0.f; B[14] = 0.f; B[15] = 1.f;
}

// ---------------------------------------------------------------------------
// Phase 2: WMMA overlap sweep.
// Block = 8 waves sharing one l-chunk of 32 boxes; TDM stages the 4KB of
// posed corners into LDS once; each wave owns 4 k-frames and runs 16
// V_WMMA_F32_16X16X4_F32 (A = 4 stacked inverses, B = 16 corners from LDS).
// Per-axis ballots land directly in SGPR lane-masks (abs as v_cmp modifier),
// combined with s_and; lanes 0-7 accumulate one (frame, parity) bit-stream,
// Morton-merged and stored once per wave.
// ---------------------------------------------------------------------------
__global__ void __launch_bounds__(256)
phase2_wmma_overlap(const float* __restrict__ btWS,
                    const float* __restrict__ hposedWS,
                    const float* __restrict__ bbmin,
                    const float* __restrict__ bbmax,
                    unsigned int* __restrict__ rawWS)   // K x 32 dwords (bitmask rows)
{
    __shared__ float sh[32 * 8 * 4];   // 32 boxes x 8 corners x (x,y,z,1) = 4KB

    const int lane  = threadIdx.x & 31;
    const int wave  = threadIdx.x >> 5;
    const int kbase = blockIdx.x * 32 + wave * 4;   // 4 k-frames per wave
    const int lbase = blockIdx.y * 32;              // 32 l-boxes per block
    const int half  = lane >> 4;
    const int m     = lane & 15;

    // ---- TDM: stage hposed[lbase .. lbase+32) into LDS (one DMA per block) ----
    if (threadIdx.x == 0) {
        unsigned long long gaddr =
            (unsigned long long)(const void*)(hposedWS + (size_t)lbase * 32);
        unsigned lds_off = (unsigned)(size_t)(void*)sh;   // LDS byte offset

        u32x4 g0;
        g0[0] = 1u;                                        // count=1, user D#
        g0[1] = lds_off;                                   // lds_addr
        g0[2] = (unsigned)(gaddr & 0xFFFFFFFFu);           // global_addr[31:0]
        g0[3] = (unsigned)((gaddr >> 32) & 0x01FFFFFFu)    // global_addr[56:32]
              | 0x80000000u;                               // type=2 ("image")

        i32x8 g1;
        g1[0] = (int)(2u << 16);            // data_size = 4B; no mask/flags
        g1[1] = (int)(0x1000u << 16);       // tensor_dim0 = 4096 (low 16 bits)
        g1[2] = (int)(1u << 16);            // tensor_dim0 hi=0; tensor_dim1 = 1
        g1[3] = (int)(0x1000u << 16);       // tensor_dim1 hi=0; tile_dim0 = 4096
        g1[4] = 1;                          // tile_dim1 = 1; tile_dim2 = 0
        g1[5] = 0x1000;                     // tensor_dim0_stride = 4096
        g1[6] = 0;
        g1[7] = 0;

        i32x4 gz4 = {0, 0, 0, 0};
        i32x8 gz8 = {0, 0, 0, 0, 0, 0, 0, 0};
        __builtin_amdgcn_tensor_load_to_lds(g0, g1, gz4, gz4, gz8, 0);
        __builtin_amdgcn_s_wait_tensorcnt(0);
    }
    __syncthreads();

    // ---- loop-invariant operands ----
    // A: lane holds A[m][2*half + {0,1}] of the 16x4 stacked-inverse matrix
    const float* arow = btWS + (size_t)(kbase + (m >> 2)) * 16 + (m & 3) * 4 + 2 * half;
    v2f a; a.x = arow[0]; a.y = arow[1];

    const int kA = kbase + 2 * half;   // frames tested by this lane's D rows
    const int kB = kA + 1;
    // centered interval form: inside <=> |proj - c| < h, h = halfext - eps
    const float cAx = 0.5f * (bbmin[kA*3+0] + bbmax[kA*3+0]);
    const float cAy = 0.5f * (bbmin[kA*3+1] + bbmax[kA*3+1]);
    const float cAz = 0.5f * (bbmin[kA*3+2] + bbmax[kA*3+2]);
    const float hAx = 0.5f * (bbmax[kA*3+0] - bbmin[kA*3+0]) - EPS_F;
    const float hAy = 0.5f * (bbmax[kA*3+1] - bbmin[kA*3+1]) - EPS_F;
    const float hAz = 0.5f * (bbmax[kA*3+2] - bbmin[kA*3+2]) - EPS_F;
    const float cBx = 0.5f * (bbmin[kB*3+0] + bbmax[kB*3+0]);
    const float cBy = 0.5f * (bbmin[kB*3+1] + bbmax[kB*3+1]);
    const float cBz = 0.5f * (bbmin[kB*3+2] + bbmax[kB*3+2]);
    const float hBx = 0.5f * (bbmax[kB*3+0] - bbmin[kB*3+0]) - EPS_F;
    const float hBy = 0.5f * (bbmax[kB*3+1] - bbmin[kB*3+1]) - EPS_F;
    const float hBz = 0.5f * (bbmax[kB*3+2] - bbmin[kB*3+2]) - EPS_F;

    // per-lane result-stream decode (lanes 0-7 carry real streams):
    //   frame j = lane&3, parity p = (lane>>2)&1; byte of mask = (j&2)+p
    const unsigned oddm  = (unsigned)-(int)(lane & 1);                 // j odd -> maskB
    const unsigned shift = 8u * (unsigned)((lane & 2) + ((lane >> 2) & 1));
    unsigned acc = 0;   // bit it = inside(k_{j}, box 2*it+p)

#pragma unroll 4
    for (int it = 0; it < 16; ++it) {
        const int relbox = 2 * it + (m >> 3);
        // B operand from LDS: components 2*half+{0,1} of corner (relbox, m&7)
        const float* brow = sh + (size_t)(relbox * 8 + (m & 7)) * 4 + 2 * half;
        v2f b; b.x = brow[0]; b.y = brow[1];

        v8f d = {};
        d = __builtin_amdgcn_wmma_f32_16x16x4_f32(
            false, a, false, b, (short)0, d, false, false);

        // per-axis ballots: each lowers to one v_cmp_lt_f32_e64 sN, |v|, h
        unsigned maskA =
            __builtin_amdgcn_ballot_w32(__builtin_fabsf(d[0] - cAx) < hAx) &
            __builtin_amdgcn_ballot_w32(__builtin_fabsf(d[1] - cAy) < hAy) &
            __builtin_amdgcn_ballot_w32(__builtin_fabsf(d[2] - cAz) < hAz);
        unsigned maskB =
            __builtin_amdgcn_ballot_w32(__builtin_fabsf(d[4] - cBx) < hBx) &
            __builtin_amdgcn_ballot_w32(__builtin_fabsf(d[5] - cBy) < hBy) &
            __builtin_amdgcn_ballot_w32(__builtin_fabsf(d[6] - cBz) < hBz);

        // byte groups of maskA: 0:(k0,p0) 1:(k0,p1) 2:(k2,p0) 3:(k2,p1); maskB: k1/k3
        unsigned sel   = maskA ^ ((maskA ^ maskB) & oddm);   // s_xor + v_and + v_xor
        unsigned bytev = (sel >> shift) & 0xFFu;             // v_bfe
        acc |= uminu(bytev, 1u) << it;                       // v_min + v_lshl_or
    }

    // merge parity streams: lane g<4 pairs with lane g+4 (same frame, p=1)
    unsigned accP = (unsigned)__shfl((int)acc, (lane & 3) + 4, 32);
    if (lane < 4) {
        unsigned w = spread16(acc) | (spread16(accP) << 1);   // bit l_rel set
        rawWS[(size_t)(kbase + lane) * 32 + blockIdx.y] = w;
    }
}

// ---------------------------------------------------------------------------
// Phase 3: inside = tril((raw&dis) | (raw&dis)^T), emitted as float 0/1.
// ---------------------------------------------------------------------------
__global__ void phase3_symm_tril(const unsigned int* __restrict__ rawWS,
                                 const int* __restrict__ disable,
                                 float* __restrict__ outInside)
{
    int idx = blockIdx.x * blockDim.x + threadIdx.x;
    if (idx >= KBOX * KBOX) return;
    int k = idx >> 10;
    int l = idx & (KBOX - 1);
    float v = 0.0f;
    if (l <= k) {
        bool kl = ((rawWS[(size_t)k * 32 + (l >> 5)] >> (l & 31)) & 1u) &&
                  (disable[(size_t)k * KBOX + l] != 0);
        bool lk = ((rawWS[(size_t)l * 32 + (k >> 5)] >> (k & 31)) & 1u) &&
                  (disable[(size_t)l * KBOX + k] != 0);
        v = (kl || lk) ? 1.0f : 0.0f;
    }
    outInside[idx] = v;
}

// ---------------------------------------------------------------------------
extern "C" void kernel_launch(void* const* d_in, const int* in_sizes, int n_in,
                              void* d_out, int out_size, void* d_ws, size_t ws_size,
                              hipStream_t stream)
{
    (void)in_sizes; (void)n_in; (void)out_size; (void)ws_size;

    const float* bbmin = (const float*)d_in[0];   // K x 3
    const float* bbmax = (const float*)d_in[1];   // K x 3
    const float* T     = (const float*)d_in[2];   // K x 4 x 4
    const int*   dis   = (const int*)  d_in[3];   // K x K

    float* btWS     = (float*)d_ws;                       // K*16 floats
    float* hposedWS = btWS + (size_t)KBOX * 16;           // K*8*4 floats
    unsigned int* rawWS = (unsigned int*)(hposedWS + (size_t)KBOX * 8 * 4); // K*32 dwords

    float* outPosed  = (float*)d_out;                     // K*8*3 floats
    float* outInside = outPosed + (size_t)KBOX * 8 * 3;   // K*K floats

    phase1_pose_and_invert<<<dim3(KBOX / 256), 256, 0, stream>>>(
        bbmin, bbmax, T, btWS, hposedWS, outPosed);

    // 32 k-supergroups x 32 l-chunks; 8 waves/block, 16 WMMAs/wave
    phase2_wmma_overlap<<<dim3(KBOX / 32, KBOX / 32), 256, 0, stream>>>(
        btWS, hposedWS, bbmin, bbmax, rawWS);

    phase3_symm_tril<<<dim3((KBOX * KBOX) / 256), 256, 0, stream>>>(
        rawWS, dis, outInside);
}